// FastRayTransformation_18133351923980
// MI455X (gfx1250) — compile-verified
//
#include <hip/hip_runtime.h>
#include <cstdint>

// Problem constants (match reference)
#define NXC 200
#define NYC 200
#define NZC 6
#define NUM_VOXELS (NXC * NYC * NZC)   // 240000
#define BB 4
#define NN 6
#define CC 64
#define HH 64
#define WW 176
#define HWSZ (HH * WW)                 // 11264
#define CHUNK 16

// out[b, c, vox] = valid[vox] ? feat[b, cam[vox], c, v[vox], u[vox]] : 0
//
// lane -> voxel (coalesced index loads + coalesced NT stores), blockIdx.y -> b.
// Validity branch hoisted to thread scope so the valid path issues clean
// clauses of 16 unconditional gathers (max MLP, one s_wait per chunk) and the
// invalid path issues zero loads.
__global__ __launch_bounds__(256) void FastRayTransformation_gather(
    const float* __restrict__ feat,
    const int* __restrict__ cam_idx,
    const int* __restrict__ u_idx,
    const int* __restrict__ v_idx,
    const uint8_t* __restrict__ valid,
    float* __restrict__ out)
{
    const int vox = blockIdx.x * 256 + threadIdx.x;
    if (vox >= NUM_VOXELS) return;
    const int b = blockIdx.y;

    // Coalesced per-voxel index loads (lanes -> consecutive voxels)
    const int cam = cam_idx[vox];
    const int uu  = u_idx[vox];
    const int vv  = v_idx[vox];
    const bool ok = valid[vox] != 0;

    // out[b][c][vox]: channel stride = NUM_VOXELS
    float* __restrict__ dst = out + (size_t)b * CC * NUM_VOXELS + vox;

    if (ok) {
        // feat[b][cam][c][vv][uu]: base at c=0, channel stride = H*W
        const float* __restrict__ src =
            feat + (size_t)((b * NN + cam) * CC) * HWSZ + (size_t)vv * WW + uu;

        // 4 chunks: 16 unconditional RT gathers in flight (load clause),
        // one wait, then 16 fully-coalesced NT stores (store clause).
#pragma unroll
        for (int cb = 0; cb < CC; cb += CHUNK) {
            float vals[CHUNK];
#pragma unroll
            for (int i = 0; i < CHUNK; ++i) {
                vals[i] = src[(size_t)(cb + i) * HWSZ];   // RT: keep feat in L2
            }
#pragma unroll
            for (int i = 0; i < CHUNK; ++i) {
                __builtin_nontemporal_store(vals[i],
                    dst + (size_t)(cb + i) * NUM_VOXELS); // NT: don't pollute L2
            }
        }
    } else {
        // No loads at all for invalid voxels: just stream zeros.
#pragma unroll
        for (int c = 0; c < CC; ++c) {
            __builtin_nontemporal_store(0.0f, dst + (size_t)c * NUM_VOXELS);
        }
    }
}

extern "C" void kernel_launch(void* const* d_in, const int* in_sizes, int n_in,
                              void* d_out, int out_size, void* d_ws, size_t ws_size,
                              hipStream_t stream) {
    (void)in_sizes; (void)n_in; (void)out_size; (void)d_ws; (void)ws_size;

    const float*   feat    = (const float*)d_in[0];
    const int*     cam_idx = (const int*)d_in[1];
    const int*     u_idx   = (const int*)d_in[2];
    const int*     v_idx   = (const int*)d_in[3];
    const uint8_t* valid   = (const uint8_t*)d_in[4];   // jnp.bool_ -> 1 byte/elem
    float*         out     = (float*)d_out;

    dim3 grid((NUM_VOXELS + 255) / 256, BB, 1);
    dim3 block(256, 1, 1);
    FastRayTransformation_gather<<<grid, block, 0, stream>>>(
        feat, cam_idx, u_idx, v_idx, valid, out);
}